// GSModel_30803505447216
// MI455X (gfx1250) — compile-verified
//
#include <hip/hip_runtime.h>
#include <hip/hip_bf16.h>
#include <math.h>

// ---------------- problem constants (from reference) ----------------
#define NG      2048
#define IMG_H   128
#define IMG_W   128
#define NPIX    (IMG_H * IMG_W)
#define FXY     300.0f
#define PSTRIDE 12            // floats per gaussian record: u,v,ia,ib,ic,op,c0,c1,c2,z,pad,pad
#define STAGE   128           // gaussians staged into LDS per block iteration
#define STAGE_BYTES (STAGE * PSTRIDE * 4)

typedef __attribute__((ext_vector_type(2))) float v2f;
typedef __attribute__((ext_vector_type(8))) float v8f;

// ---------------- kernel 1: per-gaussian preprocessing ----------------
__global__ __launch_bounds__(256) void gs_preprocess(
    const float* __restrict__ pw,  const float* __restrict__ low,
    const float* __restrict__ high,const float* __restrict__ rop,
    const float* __restrict__ sraw,const float* __restrict__ rraw,
    const float* __restrict__ Rcw, const float* __restrict__ tcw,
    float* __restrict__ params, unsigned long long* __restrict__ keys)
{
  int g = blockIdx.x * blockDim.x + threadIdx.x;
  if (g >= NG) return;

  const float R00=Rcw[0],R01=Rcw[1],R02=Rcw[2];
  const float R10=Rcw[3],R11=Rcw[4],R12=Rcw[5];
  const float R20=Rcw[6],R21=Rcw[7],R22=Rcw[8];
  const float t0=tcw[0], t1=tcw[1], t2=tcw[2];

  const float p0=pw[g*3+0], p1=pw[g*3+1], p2=pw[g*3+2];
  // pc = Rcw @ pw + tcw
  const float pc0 = R00*p0 + R01*p1 + R02*p2 + t0;
  const float pc1 = R10*p0 + R11*p1 + R12*p2 + t1;
  const float pc2 = R20*p0 + R21*p1 + R22*p2 + t2;
  const float z    = pc2;
  const float invz = 1.0f / z;
  const float u = FXY * pc0 * invz + 0.5f * (float)IMG_W;
  const float v = FXY * pc1 * invz + 0.5f * (float)IMG_H;

  float op = 1.0f / (1.0f + expf(-rop[g]));

  // 3D covariance
  const float sx = expf(sraw[g*3+0]), sy = expf(sraw[g*3+1]), sz = expf(sraw[g*3+2]);
  float qw=rraw[g*4+0], qx=rraw[g*4+1], qy=rraw[g*4+2], qz=rraw[g*4+3];
  const float qn = sqrtf(qw*qw + qx*qx + qy*qy + qz*qz);
  qw/=qn; qx/=qn; qy/=qn; qz/=qn;
  const float G00=1.f-2.f*(qy*qy+qz*qz), G01=2.f*(qx*qy-qw*qz), G02=2.f*(qx*qz+qw*qy);
  const float G10=2.f*(qx*qy+qw*qz), G11=1.f-2.f*(qx*qx+qz*qz), G12=2.f*(qy*qz-qw*qx);
  const float G20=2.f*(qx*qz-qw*qy), G21=2.f*(qy*qz+qw*qx), G22=1.f-2.f*(qx*qx+qy*qy);
  const float M00=G00*sx, M01=G01*sy, M02=G02*sz;
  const float M10=G10*sx, M11=G11*sy, M12=G12*sz;
  const float M20=G20*sx, M21=G21*sy, M22=G22*sz;
  const float S00=M00*M00+M01*M01+M02*M02;
  const float S01=M00*M10+M01*M11+M02*M12;
  const float S02=M00*M20+M01*M21+M02*M22;
  const float S11=M10*M10+M11*M11+M12*M12;
  const float S12=M10*M20+M11*M21+M12*M22;
  const float S22=M20*M20+M21*M21+M22*M22;

  // J (2x3), T = J @ Rcw
  const float J00 = FXY*invz,            J02 = -FXY*pc0*invz*invz;
  const float J11 = FXY*invz,            J12 = -FXY*pc1*invz*invz;
  const float T00=J00*R00+J02*R20, T01=J00*R01+J02*R21, T02=J00*R02+J02*R22;
  const float T10=J11*R10+J12*R20, T11=J11*R11+J12*R21, T12=J11*R12+J12*R22;

  // cov2d = T Sigma T^T
  const float X0=S00*T00+S01*T01+S02*T02;
  const float X1=S01*T00+S11*T01+S12*T02;
  const float X2=S02*T00+S12*T01+S22*T02;
  const float Y0=S00*T10+S01*T11+S02*T12;
  const float Y1=S01*T10+S11*T11+S12*T12;
  const float Y2=S02*T10+S12*T11+S22*T12;
  const float a = T00*X0+T01*X1+T02*X2 + 0.3f;
  const float b = T10*X0+T11*X1+T12*X2;
  const float c = T10*Y0+T11*Y1+T12*Y2 + 0.3f;
  const float det = a*c - b*b;
  const float ia =  c/det, ib = -b/det, ic = a/det;

  // view direction: dirs = pw - (-(Rcw^T @ tcw))
  const float w0 = -(R00*t0 + R10*t1 + R20*t2);
  const float w1 = -(R01*t0 + R11*t1 + R21*t2);
  const float w2 = -(R02*t0 + R12*t1 + R22*t2);
  float dx=p0-w0, dy=p1-w1, dz=p2-w2;
  const float dn = sqrtf(dx*dx+dy*dy+dz*dz);
  dx/=dn; dy/=dn; dz/=dn;

  // SH basis (deg 3)
  const float xx=dx*dx, yy=dy*dy, zz=dz*dz, xy=dx*dy, yz=dy*dz, xz=dx*dz;
  float B[16];
  B[0]  =  0.28209479177387814f;
  B[1]  = -0.4886025119029199f * dy;
  B[2]  =  0.4886025119029199f * dz;
  B[3]  = -0.4886025119029199f * dx;
  B[4]  =  1.0925484305920792f * xy;
  B[5]  = -1.0925484305920792f * yz;
  B[6]  =  0.31539156525252005f * (2.f*zz - xx - yy);
  B[7]  = -1.0925484305920792f * xz;
  B[8]  =  0.5462742152960396f * (xx - yy);
  B[9]  = -0.5900435899266435f * dy * (3.f*xx - yy);
  B[10] =  2.890611442640554f  * xy * dz;
  B[11] = -0.4570457994644658f * dy * (4.f*zz - xx - yy);
  B[12] =  0.3731763325901154f * dz * (2.f*zz - 3.f*xx - 3.f*yy);
  B[13] = -0.4570457994644658f * dx * (4.f*zz - xx - yy);
  B[14] =  1.445305721320277f  * dz * (xx - yy);
  B[15] = -0.5900435899266435f * dx * (xx - 3.f*yy);

  float col[3];
#pragma unroll
  for (int ch = 0; ch < 3; ++ch) {
    float r = B[0] * low[g*3 + ch];
#pragma unroll
    for (int k = 1; k < 16; ++k) r += B[k] * high[g*45 + (k-1)*3 + ch];
    col[ch] = fmaxf(r + 0.5f, 0.0f);
  }

  // fold validity into opacity (alpha'=min(.99, 0*gw)=0 matches alpha*valid)
  if (!(z > 0.2f)) op = 0.0f;

  float* P = params + g * PSTRIDE;
  P[0]=u; P[1]=v; P[2]=ia; P[3]=ib; P[4]=ic; P[5]=op;
  P[6]=col[0]; P[7]=col[1]; P[8]=col[2]; P[9]=z; P[10]=0.f; P[11]=0.f;

  // depth-sortable key (ascending float order), index in low bits
  unsigned fu = __float_as_uint(z);
  fu ^= (fu & 0x80000000u) ? 0xFFFFFFFFu : 0x80000000u;
  keys[g] = ((unsigned long long)fu << 32) | (unsigned)g;
}

// ---------------- kernel 2: single-block bitonic depth sort ----------------
__global__ __launch_bounds__(1024) void gs_sort(
    const unsigned long long* __restrict__ keys, int* __restrict__ order)
{
  __shared__ unsigned long long sk[NG];
  const int t = threadIdx.x;
  sk[t]        = keys[t];
  sk[t + 1024] = keys[t + 1024];
  for (int k = 2; k <= NG; k <<= 1) {
    for (int j = k >> 1; j >= 1; j >>= 1) {
      __syncthreads();
      const int i = ((t & ~(j - 1)) << 1) | (t & (j - 1));
      const int p = i | j;
      const bool asc = ((i & k) == 0);
      const unsigned long long A = sk[i], Bv = sk[p];
      if ((A > Bv) == asc) { sk[i] = Bv; sk[p] = A; }
    }
  }
  __syncthreads();
  order[t]        = (int)(sk[t]        & 0xFFFFFFFFull);
  order[t + 1024] = (int)(sk[t + 1024] & 0xFFFFFFFFull);
}

// ---------------- kernel 3: gather records into sorted order ----------------
__global__ __launch_bounds__(256) void gs_gather(
    const float* __restrict__ params, const int* __restrict__ order,
    float* __restrict__ sorted)
{
  const int i = blockIdx.x * blockDim.x + threadIdx.x;
  if (i >= NG * PSTRIDE) return;
  const int g = i / PSTRIDE, f = i - g * PSTRIDE;
  sorted[i] = params[order[g] * PSTRIDE + f];
}

// ---------------- kernel 4: rasterize + blend via V_WMMA_F32_16X16X4_F32 ----
__device__ __forceinline__ float eval_alpha(const float* __restrict__ g,
                                            float xs, float ys)
{
  const float dx = g[0] - xs, dy = g[1] - ys;
  float power = -0.5f * (g[2]*dx*dx + g[4]*dy*dy) - g[3]*dx*dy;
  power = fminf(power, 0.0f);
  float al = fminf(0.99f, g[5] * __expf(power));
  return (al < (1.0f / 255.0f)) ? 0.0f : al;
}

// Issue per-lane async copies (ASYNCcnt-tracked) of one 6KB stage into LDS.
// GVS form: mem = SGPR_base + VGPR_offset ; LDS address comes from the VDST VGPR.
__device__ __forceinline__ void stage_async(const float* __restrict__ sp,
                                            unsigned ldsBase, int gBase, int tid)
{
  const unsigned gOff0 = (unsigned)(gBase * PSTRIDE * 4);
#pragma unroll
  for (int off = tid * 16; off < STAGE_BYTES; off += 256 * 16) {
    unsigned la = ldsBase + (unsigned)off;
    unsigned ga = gOff0 + (unsigned)off;
    asm volatile("global_load_async_to_lds_b128 %0, %1, %2"
                 :: "v"(la), "v"(ga), "s"(sp)
                 : "memory");
  }
}

__global__ __launch_bounds__(256) void gs_raster(
    const float* __restrict__ sp, float* __restrict__ out)
{
  __shared__ __align__(16) float lds[2][STAGE * PSTRIDE];

  const int tid  = threadIdx.x;
  const int lane = tid & 31;
  const int wave = tid >> 5;
  const int m    = lane & 15;      // pixel row within wave tile / channel column
  const int half = lane >> 4;      // 0: K=0,1  1: K=2,3 (ISA 16x4 f32 A layout)
  const int waveBase = blockIdx.x * 128 + wave * 16;
  const int pixel = waveBase + m;
  const float xs = (float)(pixel & (IMG_W - 1)) + 0.5f;
  const float ys = (float)(pixel >> 7) + 0.5f;

  // LDS byte offsets of the two staging buffers (low 32 bits of the flat LDS address)
  const unsigned ldsB0 = (unsigned)(uintptr_t)(void*)&lds[0][0];
  const unsigned ldsB1 = (unsigned)(uintptr_t)(void*)&lds[1][0];

  float Tcur = 1.0f;               // per-pixel transmittance (dup in both halves)
  v8f acc = {};                    // 16x16 f32 accumulator (pixels x channels)

  // prologue: stage 0 -> buffer 0
  stage_async(sp, ldsB0, 0, tid);
  asm volatile("s_wait_asynccnt 0" ::: "memory");
  __syncthreads();

  int cur = 0;
  for (int base = 0; base < NG; base += STAGE) {
    // kick off next stage into the idle buffer while we compute
    if (base + STAGE < NG)
      stage_async(sp, cur ? ldsB0 : ldsB1, base + STAGE, tid);
    // warm L2 for the stage after next (gfx1250 global_prefetch_b8)
    if (base + 2 * STAGE < NG && tid < STAGE_BYTES / 64)
      __builtin_prefetch(&sp[(base + 2 * STAGE) * PSTRIDE + tid * 16], 0, 3);

    const float* buf = &lds[cur][0];
    for (int c4 = 0; c4 < STAGE; c4 += 4) {
      const float* g0 = &buf[(c4 + half * 2) * PSTRIDE];
      const float* g1 = g0 + PSTRIDE;
      const float a0 = eval_alpha(g0, xs, ys);   // k = 2*half
      const float a1 = eval_alpha(g1, xs, ys);   // k = 2*half + 1
      const float q  = (1.0f - a0) * (1.0f - a1);
      const float qpart = __shfl_xor(q, 16, 32); // other half-chunk survival
      const float pre  = half ? (Tcur * qpart) : Tcur;
      const float w_lo = pre * a0;
      const float w_hi = pre * (1.0f - a0) * a1;
      Tcur *= q * qpart;                         // symmetric: both halves agree

      // B: colors of this chunk's gaussians, channel = lane&15 (>=3 -> 0)
      const int ch = (m < 3) ? m : 0;
      float b0 = g0[6 + ch];
      float b1 = g1[6 + ch];
      b0 = (m < 3) ? b0 : 0.0f;
      b1 = (m < 3) ? b1 : 0.0f;

      v2f A  = { w_lo, w_hi };                   // VGPR0 = K even, VGPR1 = K odd
      v2f Bm = { b0,  b1  };
      acc = __builtin_amdgcn_wmma_f32_16x16x4_f32(
          /*neg_a=*/false, A, /*neg_b=*/false, Bm,
          /*c_mod=*/(short)0, acc, /*reuse_a=*/false, /*reuse_b=*/false);
    }

    // next stage's async copies must land before anyone reads that buffer
    asm volatile("s_wait_asynccnt 0" ::: "memory");
    __syncthreads();
    cur ^= 1;
  }

  // D layout: lanes 0-15 hold rows 0..7 (VGPR r), lanes 16-31 rows 8..15.
  if (m < 3) {
    const int prow = waveBase + half * 8;
#pragma unroll
    for (int r = 0; r < 8; ++r)
      out[(prow + r) * 3 + m] = acc[r];
  }
}

// ---------------- host launcher ----------------
extern "C" void kernel_launch(void* const* d_in, const int* in_sizes, int n_in,
                              void* d_out, int out_size, void* d_ws, size_t ws_size,
                              hipStream_t stream)
{
  const float* pw   = (const float*)d_in[0];
  const float* low  = (const float*)d_in[1];
  const float* high = (const float*)d_in[2];
  const float* rop  = (const float*)d_in[3];
  const float* sraw = (const float*)d_in[4];
  const float* rraw = (const float*)d_in[5];
  const float* Rcw  = (const float*)d_in[6];
  const float* tcw  = (const float*)d_in[7];
  // d_in[8], d_in[9] are height/width (compile-time constants here)

  float* params = (float*)d_ws;                         // NG * PSTRIDE
  float* sorted = params + NG * PSTRIDE;                // NG * PSTRIDE
  unsigned long long* keys = (unsigned long long*)(sorted + NG * PSTRIDE); // NG
  int* order = (int*)(keys + NG);                       // NG

  gs_preprocess<<<(NG + 255) / 256, 256, 0, stream>>>(
      pw, low, high, rop, sraw, rraw, Rcw, tcw, params, keys);
  gs_sort<<<1, 1024, 0, stream>>>(keys, order);
  gs_gather<<<(NG * PSTRIDE + 255) / 256, 256, 0, stream>>>(params, order, sorted);
  gs_raster<<<NPIX / 128, 256, 0, stream>>>(sorted, (float*)d_out);
}